// BundleAdjustment_38611755991505
// MI455X (gfx1250) — compile-verified
//
#include <hip/hip_runtime.h>
#include <cstdint>

#define ACT_N 2048
#define BLOCK 256
#define IPT   8

// ---- gfx1250 async global -> LDS copy (ASYNCcnt path), inline asm so it is
// ---- portable across ROCm-7.2 clang-22 and upstream clang-23.
__device__ __forceinline__ void async_g2l_b128(void* lds_dst, const void* gsrc) {
  asm volatile("global_load_async_to_lds_b128 %0, %1, off"
               :
               : "v"((__attribute__((address_space(3))) void*)lds_dst),
                 "v"(gsrc)
               : "memory");
}

__device__ __forceinline__ void wait_async0() {
  asm volatile("s_wait_asynccnt 0" ::: "memory");
}

// Hardware transcendental sin/cos: valid here because the reference guarantees
// theta in [-0.6, 0.6] and phi in [-0.2, 0.2] rad (no argument reduction
// needed). Lowers to v_mul (1/2pi) + v_sin_f32 / v_cos_f32 TRANS ops, which
// co-issue with the surrounding VALU quaternion math.
__device__ __forceinline__ void fast_sincos(float x, float* s, float* c) {
  *s = __sinf(x);
  *c = __cosf(x);
}

// ---- Kernel 1: build normalized pose table (2048 x {t.xyz, q.xyzw, pad}) in ws.
__global__ __launch_bounds__(256) void pose_prep_kernel(
    const float* __restrict__ trans,    // (ACT_N-1, 3)
    const float* __restrict__ rot,      // (ACT_N-1, 4)
    const float* __restrict__ anchor,   // (7,)
    float4* __restrict__ poses8,        // out: ACT_N * 2 float4
    int act_n) {
  int i = blockIdx.x * blockDim.x + threadIdx.x;
  if (i >= ACT_N) return;
  float tx = 0.f, ty = 0.f, tz = 0.f, qx = 0.f, qy = 0.f, qz = 0.f, qw = 1.f;
  if (i == 0) {
    tx = anchor[0]; ty = anchor[1]; tz = anchor[2];
    qx = anchor[3]; qy = anchor[4]; qz = anchor[5]; qw = anchor[6];
  } else if (i < act_n) {
    int j = i - 1;
    tx = trans[3 * j + 0]; ty = trans[3 * j + 1]; tz = trans[3 * j + 2];
    qx = rot[4 * j + 0]; qy = rot[4 * j + 1];
    qz = rot[4 * j + 2]; qw = rot[4 * j + 3];
    float n = sqrtf(qx * qx + qy * qy + qz * qz + qw * qw);
    float inv = 1.0f / fmaxf(n, 1e-12f);
    qx *= inv; qy *= inv; qz *= inv; qw *= inv;
  }
  poses8[2 * i + 0] = make_float4(tx, ty, tz, qx);
  poses8[2 * i + 1] = make_float4(qy, qz, qw, 0.0f);
}

// ---- Kernel 2: main projection-error kernel.
__global__ __launch_bounds__(BLOCK) void project_kernel(
    const float4* __restrict__ poses8,   // ACT_N*2 float4 (in ws)
    const float2* __restrict__ pc,       // (E,2) r,theta
    const float*  __restrict__ el,       // (E,)
    const float2* __restrict__ cb,       // (E,2) baseline
    const float*  __restrict__ wgt,      // (E,)
    const float*  __restrict__ scale,    // (2,)
    const int*    __restrict__ sidx,     // (E,)
    const int*    __restrict__ tidx,     // (E,)
    const int*    __restrict__ pidx,     // (E,)
    float2*       __restrict__ out,      // (E,2)
    int n) {
  __shared__ float4 sposes[ACT_N * 2];   // 64 KB pose table

  // Stage the 64 KB pose table into LDS with async b128 copies.
  #pragma unroll
  for (int i = 0; i < (ACT_N * 2) / BLOCK; ++i) {
    int idx = i * BLOCK + threadIdx.x;
    async_g2l_b128(&sposes[idx], &poses8[idx]);
  }
  wait_async0();
  __syncthreads();

  const float s0 = scale[0];
  const float s1 = scale[1];
  const int base = blockIdx.x * (BLOCK * IPT) + threadIdx.x;

  #pragma unroll
  for (int k = 0; k < IPT; ++k) {
    int e = base + k * BLOCK;
    if (e >= n) continue;

    int p = pidx[e];
    int s = sidx[e];
    int t = tidx[e];

    float2 rt = pc[p];
    float  ph = el[p];

    float4 sA = sposes[2 * s + 0];
    float4 sB = sposes[2 * s + 1];
    float4 tA = sposes[2 * t + 0];
    float4 tB = sposes[2 * t + 1];

    // polar -> cart (r, theta, phi); |theta|<=0.6, |phi|<=0.2 -> HW trans ops.
    float sth, cth, sph, cph;
    fast_sincos(rt.y, &sth, &cth);
    fast_sincos(ph,  &sph, &cph);
    float rcp = rt.x * cph;
    float lx = rcp * cth;
    float ly = rcp * sth;
    float lz = rt.x * sph;

    // qrot(src_q, local) + src_t
    float gx, gy, gz;
    {
      float qx = sA.w, qy = sB.x, qz = sB.y, qw = sB.z;
      float t2x = 2.0f * (qy * lz - qz * ly);
      float t2y = 2.0f * (qz * lx - qx * lz);
      float t2z = 2.0f * (qx * ly - qy * lx);
      gx = lx + qw * t2x + (qy * t2z - qz * t2y) + sA.x;
      gy = ly + qw * t2y + (qz * t2x - qx * t2z) + sA.y;
      gz = lz + qw * t2z + (qx * t2y - qy * t2x) + sA.z;
    }

    // qrot(conj(tgt_q), glob - tgt_t)
    float dx = gx - tA.x, dy = gy - tA.y, dz = gz - tA.z;
    float qx = -tA.w, qy = -tB.x, qz = -tB.y, qw = tB.z;
    float t2x = 2.0f * (qy * dz - qz * dy);
    float t2y = 2.0f * (qz * dx - qx * dz);
    float t2z = 2.0f * (qx * dy - qy * dx);
    float ox = dx + qw * t2x + (qy * t2z - qz * t2y);
    float oy = dy + qw * t2y + (qz * t2x - qx * t2z);
    float oz = dz + qw * t2z + (qx * t2y - qy * t2x);

    // cart -> polar; only r and azimuth are consumed downstream.
    // atan2f stays precise: ox/oy have arbitrary signs after two rotations.
    float rr  = sqrtf(ox * ox + oy * oy + oz * oz);
    float ang = atan2f(oy, ox);

    float  w = wgt[e];
    float2 c = cb[e];
    float2 res;
    res.x = (rr  * s0 - c.x) * w;            // ERR_SCALE[0] = 1.0
    res.y = (ang * s1 - c.y) * (0.1f * w);   // ERR_SCALE[1] = 0.1
    out[e] = res;
  }
}

extern "C" void kernel_launch(void* const* d_in, const int* in_sizes, int n_in,
                              void* d_out, int out_size, void* d_ws, size_t ws_size,
                              hipStream_t stream) {
  const float*  trans  = (const float*)d_in[0];   // translation_optim (ACT_N-1,3)
  const float*  rot    = (const float*)d_in[1];   // rotation_optim    (ACT_N-1,4)
  const float*  anchor = (const float*)d_in[2];   // poses_anchor      (7,)
  const float2* pc     = (const float2*)d_in[3];  // patch_coords_r_theta (E,2)
  const float*  el     = (const float*)d_in[4];   // elevation_angle   (E,)
  const float2* cb     = (const float2*)d_in[5];  // coords_baseline   (E,2)
  const float*  wgt    = (const float*)d_in[6];   // weights           (E,)
  const float*  scale  = (const float*)d_in[7];   // physic2fls_scale  (2,)
  const int*    sidx   = (const int*)d_in[8];     // source_frame_idx  (E,)
  const int*    tidx   = (const int*)d_in[9];     // target_frame_idx  (E,)
  const int*    pidx   = (const int*)d_in[10];    // patch_idx         (E,)

  const int n     = in_sizes[8];                  // E
  int act_n       = in_sizes[0] / 3 + 1;          // ACT_N (2048)
  if (act_n > ACT_N) act_n = ACT_N;

  float4* poses8 = (float4*)d_ws;                 // 64 KB pose table in scratch

  pose_prep_kernel<<<(ACT_N + 255) / 256, 256, 0, stream>>>(
      trans, rot, anchor, poses8, act_n);

  const int grid = (n + BLOCK * IPT - 1) / (BLOCK * IPT);
  project_kernel<<<grid, BLOCK, 0, stream>>>(
      poses8, pc, el, cb, wgt, scale, sidx, tidx, pidx, (float2*)d_out, n);
}